// NetVLAD_AvgPooling_55886114455955
// MI455X (gfx1250) — compile-verified
//
#include <hip/hip_runtime.h>
#include <hip/hip_bf16.h>

// NetVLAD + AvgPooling for MI455X (gfx1250, wave32).
// B=64, M=128, D=2048, K=32.  f32 end-to-end using V_WMMA_F32_16X16X4_F32.
//
// Pipeline (all on `stream`):
//   memset   : zero gsum + asum accumulators in ws
//   avgpool_k: masked mean over raw frames                      (512 blocks)
//   assign_k : rn + logits(WMMA) + softmax + a + asum           (512 blocks)
//   axgemm_k : ax = a'^T x (WMMA, rn folded into a'), -asum*C   (2048 blocks)
//   rownorm  : intra-normalize vlad rows, accumulate gsum       (2048 blocks)
//   gnorm    : global L2 normalize                              (4096 blocks)

#define BQ 64
#define MQ 128
#define DQ 2048
#define KQ 32
#define OUTD (DQ + KQ * DQ)   // 67584 floats per batch row
#define EPSF 1e-12f

typedef float v2f __attribute__((ext_vector_type(2)));
typedef float v8f __attribute__((ext_vector_type(8)));

__device__ __forceinline__ v8f wmma_f32(v2f a, v2f b, v8f c) {
  return __builtin_amdgcn_wmma_f32_16x16x4_f32(false, a, false, b,
                                               (short)0, c, false, false);
}

// ---------------------------------------------------------------------------
// Kernel 1: masked average pooling over raw frames.  grid (D/256, B)
// ---------------------------------------------------------------------------
__global__ __launch_bounds__(256) void avgpool_k(const float* __restrict__ x,
                                                 const int* __restrict__ lengths,
                                                 float* __restrict__ out) {
  const int b = blockIdx.y;
  const int d = blockIdx.x * 256 + threadIdx.x;
  const int len = lengths[b];
  const float* xb = x + (size_t)b * MQ * DQ + d;
  float s = 0.0f;
  for (int m = 0; m < len; ++m) s += xb[(size_t)m * DQ];
  out[(size_t)b * OUTD + d] = s / (float)len;
}

// ---------------------------------------------------------------------------
// Kernel 2: one block per (b, 16-row m-tile): frame norms, logits via WMMA
// (raw x * W^T, rn folded in afterwards), softmax + ragged mask, write a,
// atomic column sums into asum.  grid (M/16, B) = (8, 64)
// ---------------------------------------------------------------------------
__global__ __launch_bounds__(256) void assign_k(const float* __restrict__ x,
                                                const int* __restrict__ lengths,
                                                const float* __restrict__ W,
                                                float* __restrict__ a_ws,
                                                float* __restrict__ rn_ws,
                                                float* __restrict__ asum_ws) {
  const int b    = blockIdx.y;
  const int mt   = blockIdx.x;             // 0..7
  const int t    = threadIdx.x;
  const int lane = t & 31;
  const int wv   = t >> 5;                 // wave id 0..7
  const int m0   = mt * 16;

  // Xs[16][260] | Ws[32][260]; later: L[16][32] aliases Xs, P[8][16][32] aliases Ws
  __shared__ float smem[4160 + 8320];
  __shared__ float rnL[16];
  float* Xs  = smem;
  float* Wsh = smem + 4160;

  const float* xb = x + (size_t)b * MQ * DQ + (size_t)m0 * DQ;
  const int len = lengths[b];

  // ---- per-frame inverse L2 norms: 2 rows per wave -------------------------
  for (int r = 0; r < 2; ++r) {
    const int row = wv * 2 + r;
    const float* xr = xb + (size_t)row * DQ;
    float s = 0.0f;
    for (int j = lane; j < DQ; j += 32) { const float v = xr[j]; s += v * v; }
    for (int off = 16; off; off >>= 1) s += __shfl_xor(s, off);
    if (lane == 0) {
      const float rv = 1.0f / fmaxf(sqrtf(s), EPSF);
      rnL[row] = rv;
      rn_ws[b * MQ + m0 + row] = rv;
    }
  }
  __syncthreads();

  // ---- logits' = x_tile * W^T (WMMA); each wave owns a 32-wide D-slice -----
  v8f acc0 = {}; v8f acc1 = {};
  const int ar   = lane & 15;              // A row within tile
  const int koff = (lane >> 4) * 2;        // K sub-offset per ISA A/B layout
  const int nidx = lane & 15;              // B column within n-tile

  for (int dc = 0; dc < DQ; dc += 256) {
#pragma unroll
    for (int i = 0; i < 16; ++i) {         // 4096 elements of raw x tile
      const int idx = t + i * 256;
      const int row = idx >> 8, col = idx & 255;
      Xs[row * 260 + col] = xb[(size_t)row * DQ + dc + col];
    }
#pragma unroll
    for (int i = 0; i < 32; ++i) {         // 8192 elements of W tile
      const int idx = t + i * 256;
      const int row = idx >> 8, col = idx & 255;
      Wsh[row * 260 + col] = W[(size_t)row * DQ + dc + col];
    }
    __syncthreads();
#pragma unroll
    for (int kk = 0; kk < 8; ++kk) {       // 8 x (16x16x4) over this wave's slice
      const int c0 = wv * 32 + kk * 4 + koff;
      v2f A;  A.x  = Xs[ar * 260 + c0];            A.y  = Xs[ar * 260 + c0 + 1];
      v2f B0; B0.x = Wsh[nidx * 260 + c0];         B0.y = Wsh[nidx * 260 + c0 + 1];
      v2f B1; B1.x = Wsh[(16 + nidx) * 260 + c0];  B1.y = Wsh[(16 + nidx) * 260 + c0 + 1];
      acc0 = wmma_f32(A, B0, acc0);
      acc1 = wmma_f32(A, B1, acc1);
    }
    __syncthreads();
  }

  // ---- spill per-wave partial tiles, reduce across 8 waves -----------------
  float* P = smem + 4160;                  // [8][16][32] (aliases Ws)
  float* L = smem;                         // [16][32]    (aliases Xs)
  {
    const int r0 = (lane >> 4) << 3;       // C/D layout: VGPR i -> row r0+i
    const int n  = lane & 15;
#pragma unroll
    for (int i = 0; i < 8; ++i) {
      P[wv * 512 + (r0 + i) * 32 + n]      = acc0[i];
      P[wv * 512 + (r0 + i) * 32 + 16 + n] = acc1[i];
    }
  }
  __syncthreads();
  for (int e = t; e < 512; e += 256) {
    float s = 0.0f;
#pragma unroll
    for (int wvi = 0; wvi < 8; ++wvi) s += P[wvi * 512 + e];
    L[e] = s;
  }
  __syncthreads();

  // ---- rn fold + softmax + mask; write a ----------------------------------
  if (t < 16) {
    const int mg = m0 + t;
    const float rv = rnL[t];
    float l[KQ];
    float mx = -3.4e38f;
#pragma unroll
    for (int k = 0; k < KQ; ++k) { l[k] = L[t * 32 + k] * rv; mx = fmaxf(mx, l[k]); }
    float s = 0.0f;
#pragma unroll
    for (int k = 0; k < KQ; ++k) { l[k] = __expf(l[k] - mx); s += l[k]; }
    const float inv = (mg < len) ? (1.0f / s) : 0.0f;
#pragma unroll
    for (int k = 0; k < KQ; ++k) {
      const float av = l[k] * inv;
      L[t * 32 + k] = av;
      a_ws[((size_t)b * MQ + mg) * KQ + k] = av;
    }
  }
  __syncthreads();
  if (t < KQ) {
    float s = 0.0f;
#pragma unroll
    for (int r = 0; r < 16; ++r) s += L[r * 32 + t];
    atomicAdd(&asum_ws[b * KQ + t], s);
  }
}

// ---------------------------------------------------------------------------
// Kernel 3: one block per (b, 64-wide d-chunk): ax = a'^T * x via WMMA with
// rn folded into a'; fused  - asum[k]*C[k,d]  on the store.
// grid (D/64, B) = (32, 64)
// ---------------------------------------------------------------------------
__global__ __launch_bounds__(256) void axgemm_k(const float* __restrict__ x,
                                                const float* __restrict__ C,
                                                const float* __restrict__ a_ws,
                                                const float* __restrict__ rn_ws,
                                                const float* __restrict__ asum_ws,
                                                float* __restrict__ out) {
  const int b    = blockIdx.y;
  const int dc   = blockIdx.x * 64;
  const int t    = threadIdx.x;
  const int lane = t & 31;
  const int wv   = t >> 5;

  __shared__ float aLDS[MQ * KQ];          // a'[m][k] = a[m][k] * rn[m]
  __shared__ float X2[MQ * 68];            // raw x chunk [128][64] (+4 pad)
  __shared__ float asumL[KQ];

  const float* xb = x + (size_t)b * MQ * DQ;

#pragma unroll
  for (int i = 0; i < 16; ++i) {           // 4096 elements of a'
    const int idx = t + i * 256;
    const int row = idx >> 5;
    aLDS[idx] = a_ws[(size_t)b * MQ * KQ + idx] * rn_ws[b * MQ + row];
  }
#pragma unroll
  for (int i = 0; i < 32; ++i) {           // 8192 elements of raw x chunk
    const int idx = t + i * 256;
    const int row = idx >> 6, col = idx & 63;
    X2[row * 68 + col] = xb[(size_t)row * DQ + dc + col];
  }
  if (t < KQ) asumL[t] = asum_ws[b * KQ + t];
  __syncthreads();

  const int kt   = wv & 1;                 // cluster-tile 0..1
  const int dt   = wv >> 1;                // d-tile 0..3
  const int r    = kt * 16 + (lane & 15);  // A row (cluster)
  const int koff = (lane >> 4) * 2;
  const int dl   = dt * 16 + (lane & 15);  // B column (d within chunk)

  v8f acc = {};
#pragma unroll
  for (int mk = 0; mk < 32; ++mk) {        // contract over M=128, 4 at a time
    const int mm = mk * 4 + koff;
    v2f A;  A.x  = aLDS[mm * KQ + r];      A.y  = aLDS[(mm + 1) * KQ + r];
    v2f Bf; Bf.x = X2[mm * 68 + dl];       Bf.y = X2[(mm + 1) * 68 + dl];
    acc = wmma_f32(A, Bf, acc);
  }

  const int row0 = kt * 16 + ((lane >> 4) << 3);
  const int dg   = dc + dt * 16 + (lane & 15);
  float* outv = out + (size_t)b * OUTD + DQ;
#pragma unroll
  for (int i = 0; i < 8; ++i) {
    const int k = row0 + i;
    outv[(size_t)k * DQ + dg] = acc[i] - asumL[k] * C[(size_t)k * DQ + dg];
  }
}

// ---------------------------------------------------------------------------
// Kernel 4: intra-normalize each (b,k) vlad row; accumulate per-b global sumsq.
// grid = B*K blocks.
// ---------------------------------------------------------------------------
__global__ __launch_bounds__(256) void vlad_rownorm(float* __restrict__ out,
                                                    float* __restrict__ gsum) {
  const int bk = blockIdx.x;
  const int b = bk >> 5, k = bk & 31;
  float* row = out + (size_t)b * OUTD + DQ + (size_t)k * DQ;
  const int t = threadIdx.x;

  float s = 0.0f;
  for (int j = t; j < DQ; j += 256) { const float v = row[j]; s += v * v; }
  for (int off = 16; off; off >>= 1) s += __shfl_xor(s, off);

  __shared__ float red[8];
  __shared__ float sinv;
  if ((t & 31) == 0) red[t >> 5] = s;
  __syncthreads();
  if (t == 0) {
    float tot = 0.0f;
    for (int i = 0; i < 8; ++i) tot += red[i];
    const float inv = 1.0f / fmaxf(sqrtf(tot), EPSF);
    sinv = inv;
    atomicAdd(&gsum[b], tot * inv * inv);    // ||row after intra-norm||^2
  }
  __syncthreads();
  const float inv = sinv;
  for (int j = t; j < DQ; j += 256) row[j] *= inv;
}

// ---------------------------------------------------------------------------
// Kernel 5: global L2 normalize each batch's flattened vlad.  grid (64, B)
// ---------------------------------------------------------------------------
__global__ __launch_bounds__(256) void vlad_gnorm(float* __restrict__ out,
                                                  const float* __restrict__ gsum) {
  const int b = blockIdx.y;
  const float inv = 1.0f / fmaxf(sqrtf(gsum[b]), EPSF);
  float4* v = reinterpret_cast<float4*>(out + (size_t)b * OUTD + DQ);
  const int i = blockIdx.x * 256 + threadIdx.x;    // 16384 float4 per batch
  float4 p = v[i];
  p.x *= inv; p.y *= inv; p.z *= inv; p.w *= inv;
  v[i] = p;
}

// ---------------------------------------------------------------------------
extern "C" void kernel_launch(void* const* d_in, const int* in_sizes, int n_in,
                              void* d_out, int out_size, void* d_ws, size_t ws_size,
                              hipStream_t stream) {
  (void)in_sizes; (void)n_in; (void)out_size; (void)ws_size;
  const float* x       = (const float*)d_in[0];
  const int*   lengths = (const int*)d_in[1];
  const float* W       = (const float*)d_in[2];
  const float* C       = (const float*)d_in[3];
  float* out = (float*)d_out;

  // ws layout (floats): gsum[64] | asum[2048] | rn[8192] | a[262144]  (~1.04 MB)
  float* gsum = (float*)d_ws;
  float* asum = gsum + BQ;
  float* rn   = asum + BQ * KQ;
  float* a    = rn + BQ * MQ;

  hipMemsetAsync(d_ws, 0, (BQ + BQ * KQ) * sizeof(float), stream);  // gsum+asum
  avgpool_k   <<<dim3(DQ / 256, BQ), 256, 0, stream>>>(x, lengths, out);
  assign_k    <<<dim3(MQ / 16, BQ),  256, 0, stream>>>(x, lengths, W, a, rn, asum);
  axgemm_k    <<<dim3(DQ / 64, BQ),  256, 0, stream>>>(x, C, a, rn, asum, out);
  vlad_rownorm<<<BQ * KQ,            256, 0, stream>>>(out, gsum);
  vlad_gnorm  <<<dim3(64, BQ),       256, 0, stream>>>(out, gsum);
}